// GCN_74036646248595
// MI455X (gfx1250) — compile-verified
//
#include <hip/hip_runtime.h>

typedef __attribute__((ext_vector_type(2))) float v2f;
typedef __attribute__((ext_vector_type(8))) float v8f;

// ---------------------------------------------------------------- utilities
__global__ void k_zero(float* __restrict__ p, long long n) {
  long long i = (long long)blockIdx.x * blockDim.x + threadIdx.x;
  if (i < n) p[i] = 0.0f;
}

// count incoming edges per node (self-loop added in k_dis)
__global__ void k_deg(const int* __restrict__ dst, float* __restrict__ deg, int E) {
  int e = blockIdx.x * blockDim.x + threadIdx.x;
  if (e < E) atomicAdd(&deg[dst[e]], 1.0f);
}

// dis[v] = rsqrt(deg[v] + 1)  (deg includes self loop -> always >= 1)
__global__ void k_dis(float* __restrict__ d, int N) {
  int v = blockIdx.x * blockDim.x + threadIdx.x;
  if (v < N) d[v] = rsqrtf(d[v] + 1.0f);
}

// -------------------------------------------------- dense GEMM via f32 WMMA
// H[Nrows x Fout] = X[Nrows x Fin] * W[Fin x Fout]
// W is staged in LDS transposed + zero-padded to Foutp columns, so the inner
// loop is branch-free: one global b64 load (A) + one LDS b64 load (B) + one
// V_WMMA_F32_16X16X4_F32 per K-step, chained through the C accumulator.
__global__ void k_gemm_wmma(const float* __restrict__ X, const float* __restrict__ W,
                            float* __restrict__ H, int Nrows, int Fin, int Fout,
                            int ntN, int tiles) {
  extern __shared__ float sWT[];               // [Foutp][Fin], transposed, padded
  const int Foutp = ntN << 4;

  // cooperative staging (all waves participate; zeros in padded columns)
  const int total = Foutp * Fin;
  for (int idx = threadIdx.x; idx < total; idx += blockDim.x) {
    int c = idx / Fin;
    int k = idx - c * Fin;
    sWT[idx] = (c < Fout) ? W[(size_t)k * Fout + c] : 0.0f;
  }
  __syncthreads();

  int wave = blockIdx.x * (blockDim.x >> 5) + (threadIdx.x >> 5);
  int lane = threadIdx.x & 31;
  if (wave >= tiles) return;                   // wave-uniform: EXEC all-1s for WMMA

  int tm = wave / ntN;
  int tn = wave - tm * ntN;

  int mrow = tm * 16 + (lane & 15);            // A-matrix row owned by this lane
  if (mrow >= Nrows) mrow = Nrows - 1;         // clamp (corrupts only unstored rows)
  int col  = tn * 16 + (lane & 15);            // B/C/D column (always < Foutp)
  int khalf = (lane >> 4) << 1;                // lanes 0-15: K+0..1, 16-31: K+2..3

  const float* __restrict__ xrow = X + (size_t)mrow * Fin + khalf;
  const float* wcol = sWT + (size_t)col * Fin + khalf;

  v8f c = {0.f, 0.f, 0.f, 0.f, 0.f, 0.f, 0.f, 0.f};
#pragma unroll 4
  for (int k0 = 0; k0 < Fin; k0 += 4) {
    v2f a = *(const v2f*)(xrow + k0);          // global_load_b64
    v2f b = *(const v2f*)(wcol + k0);          // ds_load_b64
    c = __builtin_amdgcn_wmma_f32_16x16x4_f32(false, a, false, b, (short)0, c,
                                              false, false);
  }

  // C/D layout: VGPR r -> row (tm*16 + r) lanes 0-15, (tm*16 + 8 + r) lanes 16-31
  int rbase = tm * 16 + ((lane >> 4) << 3);
  if (col < Fout) {
#pragma unroll
    for (int r = 0; r < 8; ++r) {
      int row = rbase + r;
      if (row < Nrows) H[(size_t)row * Fout + col] = c[r];
    }
  }
}

// -------------------------------------------- edge scatter: agg[dst] += H[src]*norm
__global__ void k_edge(const float* __restrict__ H, const int* __restrict__ src,
                       const int* __restrict__ dst, const float* __restrict__ dis,
                       float* __restrict__ agg, int E, int F) {
  long long tid = (long long)blockIdx.x * blockDim.x + threadIdx.x;
  int chunks = F >> 2;                         // F is a multiple of 4 (84/64/32)
  long long total = (long long)E * chunks;
  if (tid >= total) return;
  int e  = (int)(tid / chunks);
  int c4 = (int)(tid - (long long)e * chunks) << 2;
  int s = src[e], d = dst[e];
  float norm = dis[s] * dis[d];
  const float* hp = H + (size_t)s * F + c4;
  float* ap = agg + (size_t)d * F + c4;
#pragma unroll
  for (int i = 0; i < 4; ++i) atomicAdd(ap + i, hp[i] * norm);
}

// self-loop contribution + bias + ReLU (in place on agg)
__global__ void k_finish(const float* __restrict__ H, const float* __restrict__ dis,
                         const float* __restrict__ bias, float* __restrict__ agg,
                         int N, int F) {
  long long i = (long long)blockIdx.x * blockDim.x + threadIdx.x;
  if (i >= (long long)N * F) return;
  int v = (int)(i / F);
  int f = (int)(i - (long long)v * F);
  float dv = dis[v];
  float val = agg[i] + H[i] * dv * dv + bias[f];
  agg[i] = val > 0.0f ? val : 0.0f;
}

// ------------------------------------------------------------- mean pool + linear
__global__ void k_pool(const float* __restrict__ A, const int* __restrict__ batch,
                       float* __restrict__ pool, float* __restrict__ cnt,
                       int N, int F) {
  long long i = (long long)blockIdx.x * blockDim.x + threadIdx.x;
  if (i >= (long long)N * F) return;
  int v = (int)(i / F);
  int f = (int)(i - (long long)v * F);
  int g = batch[v];
  atomicAdd(&pool[(size_t)g * F + f], A[i]);
  if (f == 0) atomicAdd(&cnt[g], 1.0f);
}

__global__ void k_final(const float* __restrict__ pool, const float* __restrict__ cnt,
                        const float* __restrict__ Wl, const float* __restrict__ bl,
                        float* __restrict__ out, int G, int F, int OUT) {
  int t = blockIdx.x * blockDim.x + threadIdx.x;
  if (t >= G * OUT) return;
  int g = t / OUT;
  int o = t - g * OUT;
  float c = cnt[g];
  if (c < 1.0f) c = 1.0f;
  float s = 0.0f;
  for (int f = 0; f < F; ++f) s += pool[(size_t)g * F + f] * Wl[(size_t)f * OUT + o];
  out[t] = s / c + bl[o];
}

// ---------------------------------------------------------------------- launch
static inline int ceil_div_ll(long long a, long long b) { return (int)((a + b - 1) / b); }

extern "C" void kernel_launch(void* const* d_in, const int* in_sizes, int n_in,
                              void* d_out, int out_size, void* d_ws, size_t ws_size,
                              hipStream_t stream) {
  const float* x   = (const float*)d_in[0];
  const int*   ei  = (const int*)d_in[1];
  const int*   bat = (const int*)d_in[2];
  const float* W1  = (const float*)d_in[3];
  const float* b1  = (const float*)d_in[4];
  const float* W2  = (const float*)d_in[5];
  const float* b2  = (const float*)d_in[6];
  const float* W3  = (const float*)d_in[7];
  const float* b3  = (const float*)d_in[8];
  const float* Wl  = (const float*)d_in[9];
  const float* bl  = (const float*)d_in[10];
  float* out = (float*)d_out;

  const int N   = in_sizes[2];           // 100000
  const int Fin = in_sizes[0] / N;       // 128
  const int E   = in_sizes[1] / 2;       // 1.6M
  const int F1  = in_sizes[4];           // 84
  const int F2  = in_sizes[6];           // 64
  const int F3  = in_sizes[8];           // 32
  const int OUT = in_sizes[10];          // 2
  const int G   = 64;

  const int* src = ei;
  const int* dst = ei + E;

  float* ws   = (float*)d_ws;
  float* dis  = ws;                               // N
  float* bufH = dis + N;                          // N*F1 (reused for F2/F3)
  float* bufA = bufH + (size_t)N * F1;            // N*F1
  float* bufB = bufA + (size_t)N * F1;            // N*F2
  float* pool = bufB + (size_t)N * F2;            // G*F3
  float* cnt  = pool + (size_t)G * F3;            // G

  const int TB = 256;
  const int mt = (N + 15) / 16;

  // degree -> dis
  k_zero<<<ceil_div_ll(N, TB), TB, 0, stream>>>(dis, N);
  k_deg <<<ceil_div_ll(E, TB), TB, 0, stream>>>(dst, dis, E);
  k_dis <<<ceil_div_ll(N, TB), TB, 0, stream>>>(dis, N);

  // ---- layer 1: x(N,Fin) @ W1 -> bufH(N,F1); aggregate -> bufA
  {
    int ntN = (F1 + 15) / 16, tiles = mt * ntN;
    size_t smem = (size_t)(ntN << 4) * Fin * sizeof(float);
    k_gemm_wmma<<<ceil_div_ll(tiles, 8), 256, smem, stream>>>(x, W1, bufH, N, Fin, F1, ntN, tiles);
    k_zero<<<ceil_div_ll((long long)N * F1, TB), TB, 0, stream>>>(bufA, (long long)N * F1);
    k_edge<<<ceil_div_ll((long long)E * (F1 / 4), TB), TB, 0, stream>>>(bufH, src, dst, dis, bufA, E, F1);
    k_finish<<<ceil_div_ll((long long)N * F1, TB), TB, 0, stream>>>(bufH, dis, b1, bufA, N, F1);
  }
  // ---- layer 2: bufA(N,F1) @ W2 -> bufH(N,F2); aggregate -> bufB
  {
    int ntN = (F2 + 15) / 16, tiles = mt * ntN;
    size_t smem = (size_t)(ntN << 4) * F1 * sizeof(float);
    k_gemm_wmma<<<ceil_div_ll(tiles, 8), 256, smem, stream>>>(bufA, W2, bufH, N, F1, F2, ntN, tiles);
    k_zero<<<ceil_div_ll((long long)N * F2, TB), TB, 0, stream>>>(bufB, (long long)N * F2);
    k_edge<<<ceil_div_ll((long long)E * (F2 / 4), TB), TB, 0, stream>>>(bufH, src, dst, dis, bufB, E, F2);
    k_finish<<<ceil_div_ll((long long)N * F2, TB), TB, 0, stream>>>(bufH, dis, b2, bufB, N, F2);
  }
  // ---- layer 3: bufB(N,F2) @ W3 -> bufH(N,F3); aggregate -> bufA (reused, N*F3)
  {
    int ntN = (F3 + 15) / 16, tiles = mt * ntN;
    size_t smem = (size_t)(ntN << 4) * F2 * sizeof(float);
    k_gemm_wmma<<<ceil_div_ll(tiles, 8), 256, smem, stream>>>(bufB, W3, bufH, N, F2, F3, ntN, tiles);
    k_zero<<<ceil_div_ll((long long)N * F3, TB), TB, 0, stream>>>(bufA, (long long)N * F3);
    k_edge<<<ceil_div_ll((long long)E * (F3 / 4), TB), TB, 0, stream>>>(bufH, src, dst, dis, bufA, E, F3);
    k_finish<<<ceil_div_ll((long long)N * F3, TB), TB, 0, stream>>>(bufH, dis, b3, bufA, N, F3);
  }

  // ---- mean pool + final linear
  k_zero<<<ceil_div_ll((long long)G * F3 + G, TB), TB, 0, stream>>>(pool, (long long)G * F3 + G);
  k_pool<<<ceil_div_ll((long long)N * F3, TB), TB, 0, stream>>>(bufA, bat, pool, cnt, N, F3);
  k_final<<<1, 128, 0, stream>>>(pool, cnt, Wl, bl, out, G, F3, OUT);
}